// GAT_82952998355079
// MI455X (gfx1250) — compile-verified
//
#include <hip/hip_runtime.h>
#include <hip/hip_bf16.h>

// GAT forward on MI455X (gfx1250, wave32).
// N=100000 nodes, E=800000 edges, IN=256, H=4 heads, D=32 (H*D=128).

typedef __attribute__((ext_vector_type(2))) float v2f;
typedef __attribute__((ext_vector_type(8))) float v8f;

#define HD 128          // H*D
#define NHEADS 4
#define DDIM 32
#define SLOPE 0.2f

// ---- monotonic float<->uint encoding for atomicMax on floats ----
__device__ __forceinline__ unsigned f2o(float f) {
  unsigned u = __float_as_uint(f);
  return (u & 0x80000000u) ? ~u : (u | 0x80000000u);
}
__device__ __forceinline__ float o2f(unsigned o) {
  return __uint_as_float((o & 0x80000000u) ? (o ^ 0x80000000u) : ~o);
}
// f2o(-inf) = ~0xFF800000 = 0x007FFFFF
#define ORDERED_NEG_INF 0x007FFFFFu

// =====================================================================
// GEMM: OUT[M,128] = X[M,K] @ W[K,128], f32, via v_wmma_f32_16x16x4_f32.
// One wave handles a 16-row x 128-col panel (8 n-tiles), K-loop step 4.
// A layout (16x4 f32): lane<16 -> K=k0,k0+1 ; lane>=16 -> K=k0+2,k0+3
// B layout (4x16 f32): mirrors A (lane-half K offset = 2), N = lane&15
// C/D layout: vgpr v -> row v (lanes 0-15) / row v+8 (lanes 16-31)
// Tail rows are CLAMPED (not predicated): A row r only feeds C row r, so
// duplicated tail data is harmless; only the stores are bounds-guarded.
// This keeps the K-loop branch-free: one b64 A load + 16 b32 B loads + 8 WMMA.
// =====================================================================
__global__ __launch_bounds__(256) void gemm_wmma_f32(
    const float* __restrict__ X, const float* __restrict__ W,
    float* __restrict__ OUT, int M, int K) {
  int wave = (blockIdx.x * blockDim.x + threadIdx.x) >> 5;
  int lane = threadIdx.x & 31;
  int row0 = wave * 16;
  if (row0 >= M) return;

  int r = lane & 15;                 // row-in-tile for A, col-in-tile for B/C
  int khalf = (lane >> 4) << 1;      // 0 or 2
  int row = row0 + r;
  if (row >= M) row = M - 1;         // clamp: harmless duplicate, never stored
  const float* xrow = X + (size_t)row * K;

  v8f acc[8];
  v8f z = {0.f, 0.f, 0.f, 0.f, 0.f, 0.f, 0.f, 0.f};
#pragma unroll
  for (int t = 0; t < 8; ++t) acc[t] = z;

  for (int k0 = 0; k0 < K; k0 += 4) {
    // rows are 8B-aligned (K multiple of 4) and k0+khalf is even -> b64 load
    v2f a = *(const v2f*)(xrow + k0 + khalf);
    const float* wr = W + (size_t)(k0 + khalf) * HD;
#pragma unroll
    for (int t = 0; t < 8; ++t) {
      int col = t * 16 + r;
      v2f b;
      b.x = wr[col];
      b.y = wr[HD + col];
      acc[t] = __builtin_amdgcn_wmma_f32_16x16x4_f32(
          false, a, false, b, (short)0, acc[t], false, false);
    }
  }

  int voff = (lane >> 4) << 3;  // 0 or 8
#pragma unroll
  for (int t = 0; t < 8; ++t) {
#pragma unroll
    for (int v = 0; v < 8; ++v) {
      int rr = row0 + v + voff;
      if (rr < M) OUT[(size_t)rr * HD + t * 16 + r] = acc[t][v];
    }
  }
}

// el[i,h] = sum_d feat[i,h,d]*al[h,d] ; er likewise. One thread per (node,h).
__global__ void node_scores(const float* __restrict__ feat,
                            const float* __restrict__ al,
                            const float* __restrict__ ar,
                            float* __restrict__ el, float* __restrict__ er,
                            int n) {
  int i = blockIdx.x * blockDim.x + threadIdx.x;
  if (i >= n * NHEADS) return;
  int node = i >> 2, h = i & 3;
  const float* f = feat + (size_t)node * HD + h * DDIM;
  const float* alh = al + h * DDIM;
  const float* arh = ar + h * DDIM;
  float sl = 0.f, sr = 0.f;
#pragma unroll 8
  for (int d = 0; d < DDIM; ++d) {
    float v = f[d];
    sl += v * alh[d];
    sr += v * arh[d];
  }
  el[i] = sl;
  er[i] = sr;
}

__global__ void fill_f32(float* p, float v, long long n) {
  long long i = blockIdx.x * (long long)blockDim.x + threadIdx.x;
  if (i < n) p[i] = v;
}
__global__ void fill_u32(unsigned* p, unsigned v, long long n) {
  long long i = blockIdx.x * (long long)blockDim.x + threadIdx.x;
  if (i < n) p[i] = v;
}

// e[e,h] = leaky_relu(el[src]+er[dst]); atten[e] = mean_h e; emax via atomicMax
__global__ void edge_scores(const int* __restrict__ src,
                            const int* __restrict__ dst,
                            const float* __restrict__ el,
                            const float* __restrict__ er,
                            float* __restrict__ e_out,
                            float* __restrict__ atten_out,
                            unsigned* __restrict__ emax, int ne) {
  int e = blockIdx.x * blockDim.x + threadIdx.x;
  if (e >= ne) return;
  int s = src[e], d = dst[e];
  float sum = 0.f;
#pragma unroll
  for (int h = 0; h < NHEADS; ++h) {
    float v = el[s * NHEADS + h] + er[d * NHEADS + h];
    v = (v > 0.f) ? v : SLOPE * v;
    e_out[(size_t)e * NHEADS + h] = v;
    sum += v;
    atomicMax(&emax[d * NHEADS + h], f2o(v));
  }
  atten_out[e] = sum * (1.f / NHEADS);
}

// denom[dst,h] += exp(e - emax[dst,h])
__global__ void edge_denom(const int* __restrict__ dst,
                           const float* __restrict__ ebuf,
                           const unsigned* __restrict__ emax,
                           float* __restrict__ denom, int ne) {
  int e = blockIdx.x * blockDim.x + threadIdx.x;
  if (e >= ne) return;
  int d = dst[e];
#pragma unroll
  for (int h = 0; h < NHEADS; ++h) {
    float m = o2f(emax[d * NHEADS + h]);
    atomicAdd(&denom[d * NHEADS + h], __expf(ebuf[(size_t)e * NHEADS + h] - m));
  }
}

// accum[dst, t] += feat[src, t] * a(e, t/32)   (128 threads per edge)
__global__ __launch_bounds__(256) void aggregate(
    const int* __restrict__ src, const int* __restrict__ dst,
    const float* __restrict__ ebuf, const unsigned* __restrict__ emax,
    const float* __restrict__ denom, const float* __restrict__ feat,
    float* __restrict__ accum, int ne) {
  long long gid = blockIdx.x * (long long)blockDim.x + threadIdx.x;
  int e = (int)(gid >> 7);
  int t = (int)(gid & 127);
  if (e >= ne) return;
  int s = src[e], d = dst[e];
  int h = t >> 5;
  float m = o2f(emax[d * NHEADS + h]);
  float a = __expf(ebuf[(size_t)e * NHEADS + h] - m) / denom[d * NHEADS + h];
  float val = feat[(size_t)s * HD + t] * a;
  atomicAdd(&accum[(size_t)d * HD + t], val);
}

// h1 = elu(accum), in place
__global__ void elu_inplace(float* __restrict__ p, long long n) {
  long long i = blockIdx.x * (long long)blockDim.x + threadIdx.x;
  if (i >= n) return;
  float v = p[i];
  p[i] = (v > 0.f) ? v : (__expf(v) - 1.f);
}

// h2 = elu(rst1 + h1); write temp (in place in d_out) and per-head views
__global__ void finalize_layer1(const float* __restrict__ h1,
                                float* __restrict__ out_temp,
                                float* __restrict__ out_heads, int n) {
  long long i = blockIdx.x * (long long)blockDim.x + threadIdx.x;
  if (i >= (long long)n * HD) return;
  float v = out_temp[i] + h1[i];
  v = (v > 0.f) ? v : (__expf(v) - 1.f);
  out_temp[i] = v;
  int node = (int)(i >> 7);
  int t = (int)(i & 127);
  int h = t >> 5, d = t & 31;
  out_heads[(size_t)h * n * DDIM + (size_t)node * DDIM + d] = v;
}

static inline int cdiv(long long a, long long b) { return (int)((a + b - 1) / b); }

extern "C" void kernel_launch(void* const* d_in, const int* in_sizes, int n_in,
                              void* d_out, int out_size, void* d_ws,
                              size_t ws_size, hipStream_t stream) {
  (void)n_in; (void)out_size; (void)ws_size;
  const float* x   = (const float*)d_in[0];
  const int*   ei  = (const int*)d_in[1];
  const float* W0  = (const float*)d_in[2];
  const float* al0 = (const float*)d_in[3];
  const float* ar0 = (const float*)d_in[4];
  const float* W1  = (const float*)d_in[5];
  const float* al1 = (const float*)d_in[6];
  const float* ar1 = (const float*)d_in[7];

  const int N = in_sizes[0] / 256;  // IN = 256
  const int E = in_sizes[1] / 2;
  const int IN = 256;
  const int* src = ei;
  const int* dst = ei + E;

  // workspace layout (floats)
  float* ws = (float*)d_ws;
  float* bufA = ws;                       // [N,128] feat0 / feat1
  float* bufB = bufA + (size_t)N * HD;    // [N,128] rst0 accum -> h1
  float* e0   = bufB + (size_t)N * HD;    // [E,4]
  float* el   = e0 + (size_t)E * NHEADS;  // [N,4]
  float* er   = el + (size_t)N * NHEADS;  // [N,4]
  unsigned* emax = (unsigned*)(er + (size_t)N * NHEADS);  // [N,4]
  float* denom = (float*)(emax + (size_t)N * NHEADS);     // [N,4]

  // output layout
  float* out = (float*)d_out;
  float* out_temp  = out;                               // [N,128]
  float* out_heads = out + (size_t)N * HD;              // 4 x [N,32]
  float* out_e1    = out + (size_t)2 * N * HD;          // [E,4]
  float* out_at0   = out_e1 + (size_t)E * NHEADS;       // [E]
  float* out_at1   = out_at0 + E;                       // [E]

  const int gemm_waves0 = cdiv(N, 16);
  const int gemm_blocks = cdiv(gemm_waves0, 8);  // 8 waves/block
  const long long nhd = (long long)N * HD;

  // ---------------- layer 0 ----------------
  gemm_wmma_f32<<<gemm_blocks, 256, 0, stream>>>(x, W0, bufA, N, IN);
  node_scores<<<cdiv((long long)N * NHEADS, 256), 256, 0, stream>>>(
      bufA, al0, ar0, el, er, N);
  fill_u32<<<cdiv((long long)N * NHEADS, 256), 256, 0, stream>>>(
      emax, ORDERED_NEG_INF, (long long)N * NHEADS);
  fill_f32<<<cdiv((long long)N * NHEADS, 256), 256, 0, stream>>>(
      denom, 0.f, (long long)N * NHEADS);
  fill_f32<<<cdiv(nhd, 256), 256, 0, stream>>>(bufB, 0.f, nhd);
  edge_scores<<<cdiv(E, 256), 256, 0, stream>>>(src, dst, el, er, e0, out_at0,
                                                emax, E);
  edge_denom<<<cdiv(E, 256), 256, 0, stream>>>(dst, e0, emax, denom, E);
  aggregate<<<cdiv((long long)E * HD, 256), 256, 0, stream>>>(
      src, dst, e0, emax, denom, bufA, bufB, E);
  elu_inplace<<<cdiv(nhd, 256), 256, 0, stream>>>(bufB, nhd);  // bufB = h1

  // ---------------- layer 1 ----------------
  gemm_wmma_f32<<<gemm_blocks, 256, 0, stream>>>(bufB, W1, bufA, N, HD);
  node_scores<<<cdiv((long long)N * NHEADS, 256), 256, 0, stream>>>(
      bufA, al1, ar1, el, er, N);
  fill_u32<<<cdiv((long long)N * NHEADS, 256), 256, 0, stream>>>(
      emax, ORDERED_NEG_INF, (long long)N * NHEADS);
  fill_f32<<<cdiv((long long)N * NHEADS, 256), 256, 0, stream>>>(
      denom, 0.f, (long long)N * NHEADS);
  fill_f32<<<cdiv(nhd, 256), 256, 0, stream>>>(out_temp, 0.f, nhd);
  edge_scores<<<cdiv(E, 256), 256, 0, stream>>>(src, dst, el, er, out_e1,
                                                out_at1, emax, E);
  edge_denom<<<cdiv(E, 256), 256, 0, stream>>>(dst, out_e1, emax, denom, E);
  aggregate<<<cdiv((long long)E * HD, 256), 256, 0, stream>>>(
      src, dst, out_e1, emax, denom, bufA, out_temp, E);
  finalize_layer1<<<cdiv(nhd, 256), 256, 0, stream>>>(bufB, out_temp,
                                                      out_heads, N);
}